// DWT_2D_2662879723835
// MI455X (gfx1250) — compile-verified
//
#include <hip/hip_runtime.h>

// 2D Haar DWT, depthwise stride-2 kernel-2.
// Input  x: [8, 96, 256, 256] f32  (d_in[0])
// Filters w_ll/w_lh/w_hl/w_hh: [1,1,2,2] f32 (d_in[1..4])
// Output: [8, 384, 128, 128] f32, channel = k*96 + c
//
// Memory-bound (AI ~0.87 flop/B; ~403MB traffic => ~17us floor at 23.3TB/s).
// Strategy: the input is consumed as one linear stream (output row r uses
// input floats [512r, 512r+512)), so we stream it with double-buffered
// GLOBAL_LOAD_ASYNC_TO_LDS_B128 (CDNA5 async path, ASYNCcnt), compute the
// 2x2 patch contractions from LDS, and emit non-temporal stores. Both
// streams are single-use and larger than the 192MB L2, so loads and stores
// carry NT temporal hints.

typedef float v4f __attribute__((ext_vector_type(4)));
typedef float v2f __attribute__((ext_vector_type(2)));

#define B_DIM 8
#define C_DIM 96
#define H_DIM 256
#define W_DIM 256
#define OH (H_DIM / 2)          // 128
#define OW (W_DIM / 2)          // 128
#define OUT_ROWS (B_DIM * C_DIM * OH)        // 98304
#define ROWS_PER_TILE 4
#define TILES_TOTAL (OUT_ROWS / ROWS_PER_TILE)   // 24576
#define GRID_BLOCKS 3072
#define TILE_FLOATS (ROWS_PER_TILE * 2 * W_DIM)  // 2048 floats = 8KB
#define TILE_BYTES (TILE_FLOATS * 4)             // 8192
#define BAND_STRIDE (C_DIM * OH * OW)            // 96*128*128 = 1572864

__global__ __launch_bounds__(256) void dwt2d_haar_kernel(
    const float* __restrict__ x,
    const float* __restrict__ w_ll, const float* __restrict__ w_lh,
    const float* __restrict__ w_hl, const float* __restrict__ w_hh,
    float* __restrict__ out)
{
    __shared__ __attribute__((aligned(16))) float lds[2 * TILE_FLOATS]; // 16KB, double buffer

    const int tid = threadIdx.x;
    // Low 32 bits of a generic pointer to LDS == LDS byte address (ISA 10.2).
    const unsigned lbase =
        (unsigned)(unsigned long long)(const void*)&lds[0];

    // Filter taps (use actual input values for bit-faithfulness):
    // f = {w(0,0), w(0,1), w(1,0), w(1,1)} applied to {x00, x01, x10, x11}.
    const v4f fll = *(const v4f*)w_ll;
    const v4f flh = *(const v4f*)w_lh;
    const v4f fhl = *(const v4f*)w_hl;
    const v4f fhh = *(const v4f*)w_hh;

    const int lr = tid >> 6;   // local output row in tile: 0..3
    const int pp = tid & 63;   // output pixel pair: covers w = 2*pp, 2*pp+1

    int t = blockIdx.x;
    int cur = 0;

    // Prologue: async-copy first tile into buffer 0 (512B per wave per op).
    {
        const char* g = (const char*)(x + (size_t)t * TILE_FLOATS) + tid * 16;
        unsigned l = lbase + tid * 16;
        asm volatile("global_load_async_to_lds_b128 %0, %1, off th:TH_LOAD_NT"
                     :: "v"(l), "v"(g) : "memory");
        asm volatile("global_load_async_to_lds_b128 %0, %1, off offset:4096 th:TH_LOAD_NT"
                     :: "v"(l), "v"(g) : "memory");
    }

    for (; t < TILES_TOTAL; t += GRID_BLOCKS) {
        const int tn = t + GRID_BLOCKS;
        if (tn < TILES_TOTAL) {
            // Prefetch next tile into the other buffer (2 async ops / wave).
            const char* g = (const char*)(x + (size_t)tn * TILE_FLOATS) + tid * 16;
            unsigned l = lbase + (cur ^ 1) * TILE_BYTES + tid * 16;
            asm volatile("global_load_async_to_lds_b128 %0, %1, off th:TH_LOAD_NT"
                         :: "v"(l), "v"(g) : "memory");
            asm volatile("global_load_async_to_lds_b128 %0, %1, off offset:4096 th:TH_LOAD_NT"
                         :: "v"(l), "v"(g) : "memory");
            // Async loads complete in order: <=2 outstanding means the
            // current tile's pair has landed in LDS.
            asm volatile("s_wait_asynccnt 2" ::: "memory");
        } else {
            asm volatile("s_wait_asynccnt 0" ::: "memory");
        }
        __syncthreads();   // whole tile visible to all waves

        // ---- compute 2 output pixels x 4 bands from LDS ----
        const int base = cur * TILE_FLOATS + lr * (2 * W_DIM) + pp * 4;
        const v4f top = *(const v4f*)&lds[base];            // row 2h, cols 4pp..4pp+3
        const v4f bot = *(const v4f*)&lds[base + W_DIM];    // row 2h+1

        const int r = t * ROWS_PER_TILE + lr;   // global output row
        const int p = r >> 7;                   // plane b*96+c
        const int h = r & (OH - 1);
        const int b = p / C_DIM;
        const int c = p - b * C_DIM;
        const int obase = ((b * (4 * C_DIM) + c) << 14) + (h << 7) + (pp << 1);

        v2f oll, olh, ohl, ohh;
        #pragma unroll
        for (int j = 0; j < 2; ++j) {
            const float x00 = top[2 * j], x01 = top[2 * j + 1];
            const float x10 = bot[2 * j], x11 = bot[2 * j + 1];
            oll[j] = fll.x * x00 + fll.y * x01 + fll.z * x10 + fll.w * x11;
            olh[j] = flh.x * x00 + flh.y * x01 + flh.z * x10 + flh.w * x11;
            ohl[j] = fhl.x * x00 + fhl.y * x01 + fhl.z * x10 + fhl.w * x11;
            ohh[j] = fhh.x * x00 + fhh.y * x01 + fhh.z * x10 + fhh.w * x11;
        }

        // Streaming (non-temporal) stores: each wave's 32x8B per band is one
        // contiguous 256B line.
        __builtin_nontemporal_store(oll, (v2f*)(out + obase));
        __builtin_nontemporal_store(olh, (v2f*)(out + obase + BAND_STRIDE));
        __builtin_nontemporal_store(ohl, (v2f*)(out + obase + 2 * BAND_STRIDE));
        __builtin_nontemporal_store(ohh, (v2f*)(out + obase + 3 * BAND_STRIDE));

        __syncthreads();   // all reads of `cur` done before it is refilled
        cur ^= 1;
    }
}

extern "C" void kernel_launch(void* const* d_in, const int* in_sizes, int n_in,
                              void* d_out, int out_size, void* d_ws, size_t ws_size,
                              hipStream_t stream) {
    const float* x    = (const float*)d_in[0];
    const float* w_ll = (const float*)d_in[1];
    const float* w_lh = (const float*)d_in[2];
    const float* w_hl = (const float*)d_in[3];
    const float* w_hh = (const float*)d_in[4];
    float* out = (float*)d_out;
    (void)in_sizes; (void)n_in; (void)out_size; (void)d_ws; (void)ws_size;

    dwt2d_haar_kernel<<<GRID_BLOCKS, 256, 0, stream>>>(x, w_ll, w_lh, w_hl, w_hh, out);
}